// AttentionLayer_28166395527355
// MI455X (gfx1250) — compile-verified
//
#include <hip/hip_runtime.h>
#include <math.h>

typedef __attribute__((ext_vector_type(2))) float v2f;
typedef __attribute__((ext_vector_type(4))) float v4f;
typedef __attribute__((ext_vector_type(8))) float v8f;

#define BATCH 256
#define T_    2048
#define F_    128
#define U_    48
#define NWAVE 8
#define TILE_T 16
#define TILES_PER_WAVE (T_ / (NWAVE * TILE_T))  // 16

__device__ __forceinline__ float fast_tanh(float a) {
#if defined(__has_builtin) && __has_builtin(__builtin_amdgcn_tanhf)
    return __builtin_amdgcn_tanhf(a);
#else
    float r;
    asm("v_tanh_f32 %0, %1" : "=v"(r) : "v"(a));
    return r;
#endif
}

__global__ __launch_bounds__(256) void attn_fused_kernel(
    const float* __restrict__ x,   // (B, T, F)
    const float* __restrict__ Wm,  // (F, U)
    const float* __restrict__ bias,// (U)
    const float* __restrict__ v,   // (U, 1)
    float* __restrict__ out)       // (B, F)
{
    // W staged pre-paired: Wl2[i][u] = { W[2i][u], W[2i+1][u] }, i in [0,64), u in [0,48)
    // -> each WMMA B fragment is ONE aligned ds_load_b64 into an even VGPR pair.
    __shared__ v2f   Wl2[(F_ / 2) * U_];   // 24 KB
    __shared__ float accbuf[NWAVE * F_];   // 4 KB
    __shared__ float mdbuf[NWAVE * 2];
    __shared__ float wexp[NWAVE][TILE_T];

    const int tid   = threadIdx.x;
    const int wave  = tid >> 5;
    const int lane  = tid & 31;
    const int half  = lane >> 4;   // 0: K0/K1 side, 1: K2/K3 side
    const int l16   = lane & 15;
    const int batch = blockIdx.x;

    // Stage W into LDS as adjacent-f pairs (one-time 24 KB)
    for (int i = tid; i < (F_ / 2) * U_; i += 256) {
        const int fi = i / U_;         // pair index: covers f = 2*fi, 2*fi+1
        const int u  = i - fi * U_;
        v2f p; p.x = Wm[(2 * fi) * U_ + u]; p.y = Wm[(2 * fi + 1) * U_ + u];
        Wl2[i] = p;
    }
    __syncthreads();

    // Per-lane column constants: this lane owns column u = 16*n + l16 of C
    float biasv[3], vv[3];
#pragma unroll
    for (int n = 0; n < 3; ++n) {
        biasv[n] = bias[16 * n + l16];
        vv[n]    = v[16 * n + l16];
    }

    const float* xb = x + (size_t)batch * T_ * F_;

    float m = -__builtin_inff();
    float d = 0.0f;
    v4f acc = {0.0f, 0.0f, 0.0f, 0.0f};   // lane owns f = 4*lane .. 4*lane+3

    for (int it = 0; it < TILES_PER_WAVE; ++it) {
        const int t0 = (wave * TILES_PER_WAVE + it) * TILE_T;

        // Prefetch next tile's rows while we compute this one (512 B/row, 4 points/row)
        if (it + 1 < TILES_PER_WAVE) {
            const int t1 = t0 + TILE_T;
            const float* pf = xb + (size_t)(t1 + l16) * F_ + half * 64;
            __builtin_prefetch(pf, 0, 0);
            __builtin_prefetch(pf + 32, 0, 0);
        }

        // ---- GEMM tile: E = X(16x128) @ W(128x48) via 32 K-steps of wmma f32 16x16x4
        v8f c0 = {}, c1 = {}, c2 = {};
        // A layout 16x4 f32: lanes 0-15 -> M=l16 hold K0,K1; lanes 16-31 hold K2,K3
        const float* arow_p = xb + (size_t)(t0 + l16) * F_ + 2 * half;
#pragma unroll 4
        for (int k = 0; k < 32; ++k) {
            v2f a = *(const v2f*)(arow_p + 4 * k);
            // B fragment base: pair index (2k + half), column l16 (+16, +32 for U-tiles)
            const v2f* bbase = &Wl2[(2 * k + half) * U_ + l16];
            v2f b0 = bbase[0];
            v2f b1 = bbase[16];
            v2f b2 = bbase[32];
            c0 = __builtin_amdgcn_wmma_f32_16x16x4_f32(false, a, false, b0, (short)0, c0, false, false);
            c1 = __builtin_amdgcn_wmma_f32_16x16x4_f32(false, a, false, b1, (short)0, c1, false, false);
            c2 = __builtin_amdgcn_wmma_f32_16x16x4_f32(false, a, false, b2, (short)0, c2, false, false);
        }

        // ---- scores: s[row] = sum_u tanh(C + b) * v   (native v_tanh_f32, no branches)
        float p[8];
#pragma unroll
        for (int j = 0; j < 8; ++j) {
            float e0 = fast_tanh(c0[j] + biasv[0]);
            float e1 = fast_tanh(c1[j] + biasv[1]);
            float e2 = fast_tanh(c2[j] + biasv[2]);
            p[j] = e0 * vv[0] + e1 * vv[1] + e2 * vv[2];
        }
        // reduce across the 16 column lanes of each half
#pragma unroll
        for (int mask = 1; mask < 16; mask <<= 1) {
#pragma unroll
            for (int j = 0; j < 8; ++j) p[j] += __shfl_xor(p[j], mask, 32);
        }
        // lanes 0-15 hold rows 0..7 in p[], lanes 16-31 hold rows 8..15

        // ---- online softmax update
        float tmax = p[0];
#pragma unroll
        for (int j = 1; j < 8; ++j) tmax = fmaxf(tmax, p[j]);
        tmax = fmaxf(tmax, __shfl_xor(tmax, 16, 32));
        const float new_m = fmaxf(m, tmax);
        const float scale = __expf(m - new_m);   // 0 on first tile (m = -inf)

        float ej[8], esum = 0.0f;
#pragma unroll
        for (int j = 0; j < 8; ++j) { ej[j] = __expf(p[j] - new_m); esum += ej[j]; }
        const float dtile = esum + __shfl_xor(esum, 16, 32);

        if (l16 == 0) {
#pragma unroll
            for (int j = 0; j < 8; ++j) wexp[wave][half * 8 + j] = ej[j];
        }
        __builtin_amdgcn_wave_barrier();  // order intra-wave LDS write->read

        d = d * scale + dtile;
        m = new_m;
        acc.x *= scale; acc.y *= scale; acc.z *= scale; acc.w *= scale;

        // ---- weighted accumulation: rows re-read from WGP$ (just touched), coalesced b128
#pragma unroll
        for (int r = 0; r < TILE_T; ++r) {
            v4f xv = *(const v4f*)(xb + (size_t)(t0 + r) * F_ + 4 * lane);
            const float wr = wexp[wave][r];
            acc.x += wr * xv.x; acc.y += wr * xv.y;
            acc.z += wr * xv.z; acc.w += wr * xv.w;
        }
    }

    // ---- merge 8 waves
    *(v4f*)&accbuf[wave * F_ + 4 * lane] = acc;
    if (lane == 0) { mdbuf[wave * 2] = m; mdbuf[wave * 2 + 1] = d; }
    __syncthreads();

    if (tid < F_) {
        float M = -__builtin_inff();
#pragma unroll
        for (int w = 0; w < NWAVE; ++w) M = fmaxf(M, mdbuf[w * 2]);
        float D = 0.0f, val = 0.0f;
#pragma unroll
        for (int w = 0; w < NWAVE; ++w) {
            const float sc = __expf(mdbuf[w * 2] - M);
            D   += mdbuf[w * 2 + 1] * sc;
            val += accbuf[w * F_ + tid] * sc;
        }
        out[(size_t)batch * F_ + tid] = val / D;
    }
}

extern "C" void kernel_launch(void* const* d_in, const int* in_sizes, int n_in,
                              void* d_out, int out_size, void* d_ws, size_t ws_size,
                              hipStream_t stream) {
    const float* x  = (const float*)d_in[0];  // (256, 2048, 128)
    const float* Wm = (const float*)d_in[1];  // (128, 48)
    const float* b  = (const float*)d_in[2];  // (48,)
    const float* v  = (const float*)d_in[3];  // (48, 1)
    float* out = (float*)d_out;               // (256, 128)

    attn_fused_kernel<<<BATCH, 256, 0, stream>>>(x, Wm, b, v, out);
}